// QuantileLayer_2379411882262
// MI455X (gfx1250) — compile-verified
//
#include <hip/hip_runtime.h>
#include <hip/hip_bf16.h>

// Quantile layer: x (B=16, H=256, W=256, C=64) f32 -> out (B, C, 16) f32.
// Exact order statistics via radix-select (digits 11/8/8/5 bits).
// One block per (b,c) slice. The Tensor Data Mover DMAs the strided slice
// (stride 256B) into LDS as TWO pipelined descriptors; pass 1 overlaps the
// second DMA. All 4 select passes run from LDS (single global pass total).

#define B_DIM   16
#define C_DIM   64
#define N_DIM   65536
#define NQ      16
#define NRANKS  32
#define NBIN1   2048            // 11-bit first digit
#define HALF    (N_DIM / 2)     // 32768 elements per TDM descriptor

typedef unsigned int uint32;
typedef uint32 __attribute__((ext_vector_type(4))) v4u;
typedef int    __attribute__((ext_vector_type(4))) v4i;
typedef int    __attribute__((ext_vector_type(8))) v8i;

#if defined(__has_builtin)
#  if __has_builtin(__builtin_amdgcn_tensor_load_to_lds)
#    define HAVE_TDM 1
#  endif
#endif
#ifndef HAVE_TDM
#  define HAVE_TDM 0
#endif
#if defined(__has_include)
#  if __has_include(<hip/amd_detail/amd_gfx1250_TDM.h>)
#    define TDM_SIX_ARGS 1     // amdgpu-toolchain (therock headers): 6-arg builtin
#  else
#    define TDM_SIX_ARGS 0     // ROCm 7.2: 5-arg builtin
#  endif
#else
#  define TDM_SIX_ARGS 0
#endif

// Monotone key over float bits (ascending order preserved).
__device__ __forceinline__ uint32 u2key(uint32 u) {
    return (u & 0x80000000u) ? ~u : (u | 0x80000000u);
}

#if HAVE_TDM
// Issue one TDM descriptor: 3-D tile (1 elem contig, 256 rows, 128 planes),
// element stride 64 floats (256 B), covering `HALF` slice elements.
__device__ __forceinline__ void tdm_load_half(const float* gptr, uint32 lds_addr) {
    unsigned long long ga = (unsigned long long)(uintptr_t)gptr;
    v4u g0;
    g0.x = 1u;                                                // count=1, user mode
    g0.y = lds_addr;                                          // lds_addr
    g0.z = (uint32)(ga & 0xFFFFFFFFull);                      // global_addr[31:0]
    g0.w = (uint32)((ga >> 32) & 0x01FFFFFFull) | (2u << 30); // addr[56:32] | type=2
    uint32 w0 = (2u << 16);                  // workgroup_mask=0, data_size=4B
    uint32 w1 = (1u << 16);                  // tensor_dim0 lo16 = 1
    uint32 w2 = (256u << 16);                // dim0 hi16=0 | tensor_dim1 lo16 = 256
    uint32 w3 = (1u << 16);                  // dim1 hi16=0 | tile_dim0 = 1
    uint32 w4 = 256u | (128u << 16);         // tile_dim1=256 | tile_dim2=128
    uint32 w5 = 64u;                         // tensor_dim0_stride lo32 = 64
    uint32 w6 = (16384u << 16);              // d0s hi16=0 | tensor_dim1_stride lo16
    uint32 w7 = 0u;                          // tensor_dim1_stride hi32 = 0
    v8i g1 = {(int)w0,(int)w1,(int)w2,(int)w3,(int)w4,(int)w5,(int)w6,(int)w7};
    v4i g2 = {128, 0, 0, 0};                 // tensor_dim2 = 128; no dim3/iterate
    v4i g3 = {0, 0, 0, 0};
#if TDM_SIX_ARGS
    v8i g4 = {0, 0, 0, 0, 0, 0, 0, 0};
    __builtin_amdgcn_tensor_load_to_lds(g0, g1, g2, g3, g4, 0);
#else
    __builtin_amdgcn_tensor_load_to_lds(g0, g1, g2, g3, 0);
#endif
}
#endif

__global__ void __launch_bounds__(256)
quantile_tdm_kernel(const float* __restrict__ x, float* __restrict__ out) {
    extern __shared__ __align__(16) uint32 stage[];   // 65536 raw f32 words (256 KB)
    __shared__ uint32 h1[NBIN1];                      // 8 KB  pass-1 hist
    __shared__ uint32 tmask[NBIN1];                   // 8 KB  digit0 -> rank bitmask
    __shared__ uint32 rh[NRANKS][256];                // 32 KB per-rank refine hists
    __shared__ uint32 prefix[NRANKS];
    __shared__ uint32 want[NRANKS];
    __shared__ float  vals[NRANKS];

    const int tid   = threadIdx.x;
    const int slice = blockIdx.x;                     // b*64 + c (channel-major)
    const int b     = slice >> 6;
    const int c     = slice & (C_DIM - 1);
    const float* __restrict__ xb = x + (size_t)b * N_DIM * C_DIM + c;
    const uint4* st4 = (const uint4*)stage;

#if HAVE_TDM
    // ---- Pipelined TDM fill: two descriptors, issued by wave 0 -------------
    if (tid < 32) {
        uint32 lds0 = (uint32)(uintptr_t)(void*)&stage[0];    // low 32b = LDS offset
        tdm_load_half(xb, lds0);
        tdm_load_half(xb + (size_t)HALF * C_DIM, lds0 + HALF * sizeof(uint32));
    }
    for (int j = tid; j < NBIN1; j += 256) h1[j] = 0u;        // overlap with DMA
    if (tid < 32) __builtin_amdgcn_s_wait_tensorcnt(1);       // descriptor 0 done
    __syncthreads();

    // ---- Pass 1a: histogram first half while second half streams in --------
    for (int i = tid; i < HALF / 4; i += 256) {
        uint4 kv = st4[i];
        atomicAdd(&h1[u2key(kv.x) >> 21], 1u);
        atomicAdd(&h1[u2key(kv.y) >> 21], 1u);
        atomicAdd(&h1[u2key(kv.z) >> 21], 1u);
        atomicAdd(&h1[u2key(kv.w) >> 21], 1u);
    }
    if (tid < 32) __builtin_amdgcn_s_wait_tensorcnt(0);       // descriptor 1 done
    __syncthreads();

    // ---- Pass 1b: histogram second half -------------------------------------
    for (int i = HALF / 4 + tid; i < N_DIM / 4; i += 256) {
        uint4 kv = st4[i];
        atomicAdd(&h1[u2key(kv.x) >> 21], 1u);
        atomicAdd(&h1[u2key(kv.y) >> 21], 1u);
        atomicAdd(&h1[u2key(kv.z) >> 21], 1u);
        atomicAdd(&h1[u2key(kv.w) >> 21], 1u);
    }
    __syncthreads();
#else
    // ---- Fallback: stage with plain loads, then histogram -------------------
    for (int j = tid; j < NBIN1; j += 256) h1[j] = 0u;
    for (int i = tid; i < N_DIM; i += 256) {
        __builtin_prefetch(xb + (size_t)(i + 1024) * C_DIM, 0, 0);
        stage[i] = __float_as_uint(xb[(size_t)i * C_DIM]);
    }
    __syncthreads();
    for (int i = tid; i < N_DIM / 4; i += 256) {
        uint4 kv = st4[i];
        atomicAdd(&h1[u2key(kv.x) >> 21], 1u);
        atomicAdd(&h1[u2key(kv.y) >> 21], 1u);
        atomicAdd(&h1[u2key(kv.z) >> 21], 1u);
        atomicAdd(&h1[u2key(kv.w) >> 21], 1u);
    }
    __syncthreads();
#endif

    // ---- Resolve digit 0 for the 32 ranks (lo/hi per quantile). ------------
    // Replicates the reference's f32 q*(N-1) floor/ceil arithmetic.
    if (tid < NRANKS) {
        int   qi   = tid >> 1;
        float q    = (float)(qi + 1) * (1.0f / 17.0f);
        float idxf = q * 65535.0f;
        uint32 rank = (tid & 1) ? (uint32)ceilf(idxf) : (uint32)floorf(idxf);
        uint32 cum = 0;
        int bin = 0;
        for (; bin < NBIN1; ++bin) {
            uint32 nc = cum + h1[bin];
            if (rank < nc) break;
            cum = nc;
        }
        prefix[tid] = (uint32)bin << 21;
        want[tid]   = rank - cum;
    }
    __syncthreads();

    // ---- Refine digits: (shift,width) = (13,8), (5,8), (0,5) ---------------
    const int SHIFTS[3] = {13, 5, 0};
    const int WIDTHS[3] = {8, 8, 5};
    for (int p = 0; p < 3; ++p) {
        const int s  = SHIFTS[p];
        const int ms = s + WIDTHS[p];        // bits above the digit must match
        for (int j = tid; j < NRANKS * 256; j += 256) ((uint32*)rh)[j] = 0u;
        for (int j = tid; j < NBIN1; j += 256) {
            uint32 m = 0;
            for (int r = 0; r < NRANKS; ++r)
                if ((prefix[r] >> 21) == (uint32)j) m |= 1u << r;
            tmask[j] = m;
        }
        __syncthreads();

        for (int i = tid; i < N_DIM / 4; i += 256) {
            uint4 kv = st4[i];
            #pragma unroll
            for (int e = 0; e < 4; ++e) {
                uint32 key = u2key(e == 0 ? kv.x : e == 1 ? kv.y : e == 2 ? kv.z : kv.w);
                uint32 m = tmask[key >> 21];
                while (m) {
                    int r = __ffs(m) - 1;
                    m &= m - 1u;
                    if (((key ^ prefix[r]) >> ms) == 0u)
                        atomicAdd(&rh[r][(key >> s) & 255u], 1u);   // ds_add_u32
                }
            }
        }
        __syncthreads();

        if (tid < NRANKS) {
            uint32 w = want[tid], cum = 0;
            int nb = 1 << WIDTHS[p];
            int bin = 0;
            for (; bin < nb; ++bin) {
                uint32 nc = cum + rh[tid][bin];
                if (w < nc) break;
                cum = nc;
            }
            prefix[tid] |= (uint32)bin << s;
            want[tid]    = w - cum;
        }
        __syncthreads();
    }

    // ---- Invert key -> float, interpolate, write (B, C, Q) -----------------
    if (tid < NRANKS) {
        uint32 k = prefix[tid];
        uint32 u = (k & 0x80000000u) ? (k ^ 0x80000000u) : ~k;
        vals[tid] = __uint_as_float(u);
    }
    __syncthreads();
    if (tid < NQ) {
        float q    = (float)(tid + 1) * (1.0f / 17.0f);
        float idxf = q * 65535.0f;
        float frac = idxf - floorf(idxf);
        float vlo = vals[2 * tid], vhi = vals[2 * tid + 1];
        out[(size_t)slice * NQ + tid] = vlo + (vhi - vlo) * frac;
    }
}

extern "C" void kernel_launch(void* const* d_in, const int* in_sizes, int n_in,
                              void* d_out, int out_size, void* d_ws, size_t ws_size,
                              hipStream_t stream) {
    (void)in_sizes; (void)n_in; (void)out_size; (void)d_ws; (void)ws_size;
    const float* x   = (const float*)d_in[0];
    float*       out = (float*)d_out;

    // 256 KB dynamic LDS stage + ~50 KB static = ~306 KB of the 320 KB/WGP.
    hipLaunchKernelGGL(quantile_tdm_kernel,
                       dim3(B_DIM * C_DIM), dim3(256),
                       (size_t)N_DIM * sizeof(unsigned int), stream,
                       x, out);
}